// network_11879879542692
// MI455X (gfx1250) — compile-verified
//
#include <hip/hip_runtime.h>
#include <math.h>

// ---------------------------------------------------------------------------
// Spectral-norm RNN + log_softmax(batch axis), fused for MI455X (gfx1250).
//
// Bandwidth-bound: ~840 MB total traffic -> ~36us floor at 23.3 TB/s.
// Parallelism capped at B=4096 recurrence lanes -> 256 wave32 16-batch tiles.
// Per step, per wave: 8x v_wmma_f32_16x16x4_f32 (K=20 input proj + K=10
// recurrence chained into one accumulator), branch-free v_tanh, LDS transpose
// for the next step's A fragments, async-to-LDS double-buffered x streaming.
// log_softmax uses the tanh bound: logsumexp(v) = 1 + log(sum exp(v-1)),
// so stats accumulate during the scan (deterministic per-block partials).
// ---------------------------------------------------------------------------

#define HID 10
#define INP 20
#define SEQ 1024
#define BATCHN 4096
#define EPSF 1e-12f

#define BLK 128                       // 4 waves
#define WAVES 4
#define TILE_B 64                     // 4 waves * 16 batch rows
#define NBLK (BATCHN / TILE_B)        // 64 blocks
#define SLAB_BYTES (TILE_B * INP * 4) // 5120 B per block per t-step

#define WS_WSN 0
#define WS_PART 128
#define WS_LADJ (WS_PART + SEQ * HID * NBLK)

typedef float v2f __attribute__((ext_vector_type(2)));
typedef float v8f __attribute__((ext_vector_type(8)));

// Generic pointers to LDS on AMDGPU carry the LDS byte offset in addr[31:0].
__device__ __forceinline__ unsigned lds_off_of(const void* p) {
  return (unsigned)(size_t)p;
}

__device__ __forceinline__ void async_copy_b64(unsigned lds_off,
                                               unsigned long long gaddr) {
  asm volatile("global_load_async_to_lds_b64 %0, %1, off"
               :: "v"(lds_off), "v"(gaddr)
               : "memory");
}

__device__ __forceinline__ void wait_async0() {
  asm volatile("s_wait_asynccnt 0x0" ::: "memory");
}

// Branch-free tanh: native v_tanh_f32 on gfx1250, else exp/rcp form that
// saturates correctly (exp(2x)->inf => 1; ->0 => -1). No libm branches.
__device__ __forceinline__ float fast_tanh(float x) {
#if __has_builtin(__builtin_amdgcn_tanhf)
  return __builtin_amdgcn_tanhf(x);
#else
  const float e = __expf(2.0f * x);
  return 1.0f - 2.0f * __builtin_amdgcn_rcpf(e + 1.0f);
#endif
}

// ---------------------------------------------------------------------------
// Kernel 0: one power iteration -> W_sn = W_hh / sigma  (100 floats, trivial)
// ---------------------------------------------------------------------------
__global__ void k_sigma(const float* __restrict__ Whh,
                        const float* __restrict__ u,
                        float* __restrict__ wsn) {
  if (threadIdx.x != 0 || blockIdx.x != 0) return;
  float v[HID], wv[HID];
  float nv = 0.f;
  for (int j = 0; j < HID; ++j) {
    float s = 0.f;
    for (int i = 0; i < HID; ++i) s += Whh[i * HID + j] * u[i];
    v[j] = s; nv += s * s;
  }
  nv = sqrtf(nv) + EPSF;
  for (int j = 0; j < HID; ++j) v[j] /= nv;
  float nw = 0.f;
  for (int i = 0; i < HID; ++i) {
    float s = 0.f;
    for (int j = 0; j < HID; ++j) s += Whh[i * HID + j] * v[j];
    wv[i] = s; nw += s * s;
  }
  const float nwn = sqrtf(nw) + EPSF;
  float sigma = 0.f;
  for (int i = 0; i < HID; ++i) sigma += (wv[i] / nwn) * wv[i];
  const float inv = 1.0f / sigma;
  for (int k = 0; k < HID * HID; ++k) wsn[k] = Whh[k] * inv;
}

// ---------------------------------------------------------------------------
// Kernel 1: fused input-projection + recurrence + raw-output + exp partials
// ---------------------------------------------------------------------------
__global__ __launch_bounds__(BLK) void k_rnn(
    const float* __restrict__ x, const float* __restrict__ Wih,
    const float* __restrict__ bih, const float* __restrict__ bhh,
    const float* __restrict__ wsn, float* __restrict__ part,
    float* __restrict__ out) {
  __shared__ __align__(16) float xbuf[2][TILE_B * INP]; // async double buffer
  __shared__ __align__(16) float htile[WAVES][16][12];  // per-wave h transpose
  __shared__ float wsum[2][WAVES][HID];                 // parity double buffer

  const int tid  = threadIdx.x;
  const int wave = tid >> 5;
  const int lane = tid & 31;
  const int np   = lane & 15;            // N (hidden) index / M row for A
  const int kb   = (lane >= 16) ? 2 : 0; // K sub-offset for A/B fragments
  const int moff = (lane >= 16) ? 8 : 0; // M offset for C/D rows
  const int b0   = blockIdx.x * TILE_B;
  const bool act = (np < HID);

  // B fragment preload: B[k][n] layout, VGPR0=K=kb, VGPR1=K=kb+1.
  // Input projection: B[k][n] = Wih[n][k]  (x @ Wih^T)
  v2f bIn[5];
#pragma unroll
  for (int c = 0; c < 5; ++c) {
    const int k0 = 4 * c + kb;
    bIn[c][0] = act ? Wih[np * INP + k0] : 0.f;
    bIn[c][1] = act ? Wih[np * INP + k0 + 1] : 0.f;
  }
  // Recurrence: B[k][n] = Wsn[n][k]  (h @ Wsn^T), zero-padded to K=12
  v2f bRec[3];
#pragma unroll
  for (int c = 0; c < 3; ++c) {
    const int k0 = 4 * c + kb;
    bRec[c][0] = (act && k0 < HID) ? wsn[np * HID + k0] : 0.f;
    bRec[c][1] = (act && (k0 + 1) < HID) ? wsn[np * HID + k0 + 1] : 0.f;
  }
  const float biasv = act ? (bih[np] + bhh[np]) : 0.f;
  const v8f cinit = {biasv, biasv, biasv, biasv, biasv, biasv, biasv, biasv};

  // h0 = 0 (cols 10,11 stay zero forever -> K padding for the rec fragments)
  for (int i = tid; i < WAVES * 16 * 12; i += BLK) ((float*)htile)[i] = 0.f;
  __syncthreads();

  // Prefetch t=0 slab (contiguous 5120 B: x[(t*B + b0)*20 ...])
  {
    const unsigned lo = lds_off_of(&xbuf[0][0]);
    const unsigned long long g =
        (unsigned long long)(const void*)(x + (size_t)b0 * INP);
#pragma unroll
    for (int op = 0; op < 5; ++op) {
      const unsigned off = op * 1024u + tid * 8u;
      async_copy_b64(lo + off, g + off);
    }
  }

  for (int t = 0; t < SEQ; ++t) {
    wait_async0();          // this wave's slab portion landed
    __syncthreads();        // everyone's portion landed
    if (t + 1 < SEQ) {      // prefetch next slab into the other buffer
      const unsigned lo = lds_off_of(&xbuf[(t + 1) & 1][0]);
      const unsigned long long g = (unsigned long long)(const void*)
          (x + ((size_t)(t + 1) * BATCHN + b0) * INP);
#pragma unroll
      for (int op = 0; op < 5; ++op) {
        const unsigned off = op * 1024u + tid * 8u;
        async_copy_b64(lo + off, g + off);
      }
    }

    // --- 8 chained WMMAs: D = x*Wih^T + h*Wsn^T + bias ---
    const float* xr = &xbuf[t & 1][(wave * 16 + np) * INP];
    v8f c = cinit;
#pragma unroll
    for (int k = 0; k < 5; ++k) {
      const float2 xv = *(const float2*)(xr + 4 * k + kb);
      const v2f a = {xv.x, xv.y};
      c = __builtin_amdgcn_wmma_f32_16x16x4_f32(false, a, false, bIn[k],
                                                (short)0, c, false, false);
    }
    const float* hr = &htile[wave][np][0];
#pragma unroll
    for (int k = 0; k < 3; ++k) {
      const float2 h2 = *(const float2*)(hr + 4 * k + kb);
      const v2f a = {h2.x, h2.y};
      c = __builtin_amdgcn_wmma_f32_16x16x4_f32(false, a, false, bRec[k],
                                                (short)0, c, false, false);
    }

    // --- branch-free tanh + exp partial (all lanes; inactive lanes benign) ---
    float hv[8];
    float es = 0.f;
#pragma unroll
    for (int r = 0; r < 8; ++r) {
      hv[r] = fast_tanh(c[r]);
      es += __expf(hv[r] - 1.0f);     // shift by tanh bound (max <= 1)
    }

    // --- single predicated region: LDS transpose + raw-output stores ---
    if (act) {
      float* op_ = out + ((size_t)t * BATCHN + b0 + wave * 16 + moff) * HID + np;
#pragma unroll
      for (int r = 0; r < 8; ++r) {
        htile[wave][moff + r][np] = hv[r];
        op_[(size_t)r * HID] = hv[r];
      }
    }

    es += __shfl_xor(es, 16, 32);     // combine M=0..7 and M=8..15 halves
    const int par = t & 1;
    if (lane < HID) wsum[par][wave][lane] = es;
    __syncthreads();
    if (wave == 0 && lane < HID) {    // deterministic per-block partial
      const float s = wsum[par][0][lane] + wsum[par][1][lane] +
                      wsum[par][2][lane] + wsum[par][3][lane];
      part[((size_t)t * HID + lane) * NBLK + blockIdx.x] = s;
    }
    // no trailing barrier: next iteration writes the other wsum parity, and
    // the same parity is only rewritten after two more top-of-loop barriers.
  }
}

// ---------------------------------------------------------------------------
// Kernel 2: reduce per-block partials -> ladj[t,h] = 1 + log(sum_b exp(v-1))
// ---------------------------------------------------------------------------
__global__ void k_reduce(const float* __restrict__ part,
                         float* __restrict__ ladj) {
  const int warp = (int)((blockIdx.x * blockDim.x + threadIdx.x) >> 5);
  const int lane = threadIdx.x & 31;
  if (warp >= SEQ * HID) return;
  const float* p = part + (size_t)warp * NBLK;
  float s = p[lane] + p[lane + 32];   // NBLK == 64
#pragma unroll
  for (int o = 16; o > 0; o >>= 1) s += __shfl_down(s, o, 32);
  if (lane == 0) ladj[warp] = 1.0f + __logf(s);
}

// ---------------------------------------------------------------------------
// Kernel 3: in-place log_softmax finalize: out -= ladj[t,h]
// ---------------------------------------------------------------------------
__global__ void k_finish(float* __restrict__ out,
                         const float* __restrict__ ladj) {
  const size_t idx = (size_t)blockIdx.x * blockDim.x + threadIdx.x;
  const size_t total = (size_t)SEQ * BATCHN * HID;
  if (idx >= total) return;
  const int h = (int)(idx % HID);
  const size_t t = idx / ((size_t)BATCHN * HID);
  out[idx] -= ladj[t * HID + h];
}

extern "C" void kernel_launch(void* const* d_in, const int* in_sizes, int n_in,
                              void* d_out, int out_size, void* d_ws,
                              size_t ws_size, hipStream_t stream) {
  (void)in_sizes; (void)n_in; (void)out_size; (void)ws_size;
  const float* x   = (const float*)d_in[0];
  const float* Wih = (const float*)d_in[1];
  const float* Whh = (const float*)d_in[2];
  const float* bih = (const float*)d_in[3];
  const float* bhh = (const float*)d_in[4];
  const float* u   = (const float*)d_in[5];
  float* out = (float*)d_out;
  float* ws  = (float*)d_ws;
  float* wsn  = ws + WS_WSN;    // 100 floats
  float* part = ws + WS_PART;   // SEQ*HID*NBLK = 655360 floats (~2.6 MB)
  float* ladj = ws + WS_LADJ;   // 10240 floats

  k_sigma<<<1, 32, 0, stream>>>(Whh, u, wsn);
  k_rnn<<<NBLK, BLK, 0, stream>>>(x, Wih, bih, bhh, wsn, part, out);
  const int rwarps = SEQ * HID; // 10240 rows, one warp each
  k_reduce<<<(rwarps * 32 + 255) / 256, 256, 0, stream>>>(part, ladj);
  const size_t total = (size_t)SEQ * BATCHN * HID;
  k_finish<<<(unsigned)((total + 255) / 256), 256, 0, stream>>>(out, ladj);
}